// VNZCALayerNorm_43688407335120
// MI455X (gfx1250) — compile-verified
//
#include <hip/hip_runtime.h>
#include <hip/hip_bf16.h>

typedef __attribute__((ext_vector_type(2))) float v2f;
typedef __attribute__((ext_vector_type(8))) float v8f;

#define BB 32
#define CC 256
#define NN 2048

// ---------------- init: zero the accumulator region ----------------
__global__ void vn_init_kernel(float* accum, int n) {
    int i = blockIdx.x * blockDim.x + threadIdx.x;
    if (i < n) accum[i] = 0.0f;
}

// ---------------- pass 1: raw moments per batch --------------------
// accum[b*9 + {0..8}] = {Sx,Sy,Sz, Pxx,Pyy,Pzz, Pxy,Pxz,Pyz}
__global__ void vn_cov_kernel(const float* __restrict__ x, float* __restrict__ accum,
                              int chanPerBlock) {
    const int b  = blockIdx.y;
    const int c0 = blockIdx.x * chanPerBlock;
    float s0 = 0.f, s1 = 0.f, s2 = 0.f;
    float p00 = 0.f, p11 = 0.f, p22 = 0.f, p01 = 0.f, p02 = 0.f, p12 = 0.f;

    for (int ci = 0; ci < chanPerBlock; ++ci) {
        const size_t base = ((size_t)(b * CC + (c0 + ci))) * 3 * NN;
        const float4* X4 = (const float4*)(x + base);
        const float4* Y4 = (const float4*)(x + base + NN);
        const float4* Z4 = (const float4*)(x + base + 2 * NN);
        for (int n = threadIdx.x; n < NN / 4; n += blockDim.x) {
            float4 xv = X4[n];
            float4 yv = Y4[n];
            float4 zv = Z4[n];
            s0 += xv.x + xv.y + xv.z + xv.w;
            s1 += yv.x + yv.y + yv.z + yv.w;
            s2 += zv.x + zv.y + zv.z + zv.w;
            p00 = fmaf(xv.x, xv.x, fmaf(xv.y, xv.y, fmaf(xv.z, xv.z, fmaf(xv.w, xv.w, p00))));
            p11 = fmaf(yv.x, yv.x, fmaf(yv.y, yv.y, fmaf(yv.z, yv.z, fmaf(yv.w, yv.w, p11))));
            p22 = fmaf(zv.x, zv.x, fmaf(zv.y, zv.y, fmaf(zv.z, zv.z, fmaf(zv.w, zv.w, p22))));
            p01 = fmaf(xv.x, yv.x, fmaf(xv.y, yv.y, fmaf(xv.z, yv.z, fmaf(xv.w, yv.w, p01))));
            p02 = fmaf(xv.x, zv.x, fmaf(xv.y, zv.y, fmaf(xv.z, zv.z, fmaf(xv.w, zv.w, p02))));
            p12 = fmaf(yv.x, zv.x, fmaf(yv.y, zv.y, fmaf(yv.z, zv.z, fmaf(yv.w, zv.w, p12))));
        }
    }

    // wave32 reduction
    #pragma unroll
    for (int off = 16; off > 0; off >>= 1) {
        s0  += __shfl_down(s0, off);  s1  += __shfl_down(s1, off);  s2  += __shfl_down(s2, off);
        p00 += __shfl_down(p00, off); p11 += __shfl_down(p11, off); p22 += __shfl_down(p22, off);
        p01 += __shfl_down(p01, off); p02 += __shfl_down(p02, off); p12 += __shfl_down(p12, off);
    }
    if ((threadIdx.x & 31) == 0) {
        float* a = accum + b * 9;
        atomicAdd(a + 0, s0);  atomicAdd(a + 1, s1);  atomicAdd(a + 2, s2);
        atomicAdd(a + 3, p00); atomicAdd(a + 4, p11); atomicAdd(a + 5, p22);
        atomicAdd(a + 6, p01); atomicAdd(a + 7, p02); atomicAdd(a + 8, p12);
    }
}

// ---------------- solve: 3x3 Jacobi eigensolve, build W and t ------
// wt[b*12 + {0..8}] = W row-major; wt[b*12 + {9..11}] = t = W*mu
__global__ void vn_solve_kernel(const float* __restrict__ accum, float* __restrict__ wt) {
    const int b = threadIdx.x;
    if (b >= BB) return;
    const float Mf = (float)(CC * NN);
    const float* a = accum + b * 9;
    float S0 = a[0], S1 = a[1], S2 = a[2];
    float mu0 = S0 / Mf, mu1 = S1 / Mf, mu2 = S2 / Mf;
    float denom = Mf + 1e-6f;

    float A[3][3];
    A[0][0] = (a[3] - S0 * mu0) / denom + 1e-5f;
    A[1][1] = (a[4] - S1 * mu1) / denom + 1e-5f;
    A[2][2] = (a[5] - S2 * mu2) / denom + 1e-5f;
    A[0][1] = A[1][0] = (a[6] - S0 * mu1) / denom;
    A[0][2] = A[2][0] = (a[7] - S0 * mu2) / denom;
    A[1][2] = A[2][1] = (a[8] - S1 * mu2) / denom;

    float V[3][3] = {{1.f, 0.f, 0.f}, {0.f, 1.f, 0.f}, {0.f, 0.f, 1.f}};
    const int pp[3] = {0, 0, 1}, qq[3] = {1, 2, 2};
    for (int it = 0; it < 30; ++it) {
        int p = pp[it % 3], q = qq[it % 3];
        float apq = A[p][q];
        if (fabsf(apq) > 1e-20f) {
            float theta = (A[q][q] - A[p][p]) / (2.0f * apq);
            float t = copysignf(1.0f, theta) / (fabsf(theta) + sqrtf(theta * theta + 1.0f));
            float cc = rsqrtf(t * t + 1.0f);
            float ss = t * cc;
            for (int k = 0; k < 3; ++k) { float akp = A[k][p], akq = A[k][q];
                A[k][p] = cc * akp - ss * akq; A[k][q] = ss * akp + cc * akq; }
            for (int k = 0; k < 3; ++k) { float apk = A[p][k], aqk = A[q][k];
                A[p][k] = cc * apk - ss * aqk; A[q][k] = ss * apk + cc * aqk; }
            for (int k = 0; k < 3; ++k) { float vkp = V[k][p], vkq = V[k][q];
                V[k][p] = cc * vkp - ss * vkq; V[k][q] = ss * vkp + cc * vkq; }
        }
    }
    float w0 = rsqrtf(fmaxf(A[0][0], 1e-5f));
    float w1 = rsqrtf(fmaxf(A[1][1], 1e-5f));
    float w2 = rsqrtf(fmaxf(A[2][2], 1e-5f));

    float* o = wt + b * 12;
    float W[3][3];
    for (int i = 0; i < 3; ++i)
        for (int j = 0; j < 3; ++j) {
            W[i][j] = V[i][0] * w0 * V[j][0] + V[i][1] * w1 * V[j][1] + V[i][2] * w2 * V[j][2];
            o[i * 3 + j] = W[i][j];
        }
    for (int i = 0; i < 3; ++i)
        o[9 + i] = W[i][0] * mu0 + W[i][1] * mu1 + W[i][2] * mu2;
}

// ---------------- pass 2: apply W via WMMA, affine-augmented -------
// D(16x16) = A(16x4) x B(4x16):
//   A rows 0..2 = W, A column K=3 = -t  (affine mean removal)
//   B rows K=0..2 = x,y,z of 16 points, row K=3 = ones
// A layout (ISA 7.12.2, f32 16x4): lanes 0-15 hold {K0,K1}, lanes 16-31 hold {K2,K3}.
// B layout assumed mirrored:       VGPR0 = {K0 | K1}, VGPR1 = {K2 | K3} by lane half.
// D layout: acc[i], lanes 0-15 = row M=i -> rows 0..2 are the whitened components.
// All loads are unconditional (no exec divergence around WMMA); lanes>=16 re-read
// the z-row cachelines that lanes 0-15 touch anyway, then select 1.0 for the K=3 row.
__global__ void vn_apply_kernel(const float* __restrict__ x, const float* __restrict__ gamma,
                                const float* __restrict__ wt, float* __restrict__ out) {
    const int b = blockIdx.y;
    const int c = blockIdx.x;
    const int lane = threadIdx.x & 31;
    const int wave = threadIdx.x >> 5;
    const int nwaves = blockDim.x >> 5;
    const bool lo = (lane < 16);

    const float* Wb = wt + b * 12;
    const int m = lane & 15;
    float ax = 0.f, ay = 0.f;
    if (m < 3) {
        if (lo) { ax = Wb[m * 3 + 0]; ay = Wb[m * 3 + 1]; }   // K=0, K=1
        else    { ax = Wb[m * 3 + 2]; ay = -Wb[9 + m]; }      // K=2, K=3 (-t)
    }
    v2f Am; Am.x = ax; Am.y = ay;

    const size_t base = ((size_t)(b * CC + c)) * 3 * NN;
    const float* X = x + base;
    float* O = out + base;
    const float g = gamma[c];
    const int rowOff = lo ? 0 : NN;   // lanes 0-15: x-row (K0) | lanes 16-31: y-row (K1)

    for (int p0 = wave * 32; p0 < NN; p0 += 32 * nwaves) {
        const int p  = p0 + (lane & 15);
        // two independent 16-point tiles in flight
        v2f B0, B1;
        B0.x = X[rowOff + p];
        B1.x = X[rowOff + p + 16];
        float z0 = X[2 * NN + p];
        float z1 = X[2 * NN + p + 16];
        B0.y = lo ? z0 : 1.0f;   // K2 = z-row | K3 = ones
        B1.y = lo ? z1 : 1.0f;

        v8f acc0 = {}, acc1 = {};
        acc0 = __builtin_amdgcn_wmma_f32_16x16x4_f32(false, Am, false, B0,
                                                     (short)0, acc0, false, false);
        acc1 = __builtin_amdgcn_wmma_f32_16x16x4_f32(false, Am, false, B1,
                                                     (short)0, acc1, false, false);
        if (lo) {
            __builtin_nontemporal_store(acc0[0] * g, O + p);
            __builtin_nontemporal_store(acc0[1] * g, O + NN + p);
            __builtin_nontemporal_store(acc0[2] * g, O + 2 * NN + p);
            __builtin_nontemporal_store(acc1[0] * g, O + p + 16);
            __builtin_nontemporal_store(acc1[1] * g, O + NN + p + 16);
            __builtin_nontemporal_store(acc1[2] * g, O + 2 * NN + p + 16);
        }
    }
}

extern "C" void kernel_launch(void* const* d_in, const int* in_sizes, int n_in,
                              void* d_out, int out_size, void* d_ws, size_t ws_size,
                              hipStream_t stream) {
    const float* x     = (const float*)d_in[0];
    const float* gamma = (const float*)d_in[1];
    float* out = (float*)d_out;

    float* accum = (float*)d_ws;        // BB*9 floats
    float* wt    = accum + BB * 9;      // BB*12 floats

    vn_init_kernel<<<dim3(1), dim3(512), 0, stream>>>(accum, BB * 9);

    const int chanPerBlock = 8;
    vn_cov_kernel<<<dim3(CC / chanPerBlock, BB), dim3(256), 0, stream>>>(x, accum, chanPerBlock);

    vn_solve_kernel<<<dim3(1), dim3(32), 0, stream>>>(accum, wt);

    vn_apply_kernel<<<dim3(CC, BB), dim3(256), 0, stream>>>(x, gamma, wt, out);
}